// ToRGB_7773890805883
// MI455X (gfx1250) — compile-verified
//
#include <hip/hip_runtime.h>

typedef __attribute__((ext_vector_type(2))) float v2f;
typedef __attribute__((ext_vector_type(8))) float v8f;

#define C_IN   512
#define HW_    128
#define NBATCH 8
#define K_RAW  9
#define K_PAD  12           // tap order t = dy*4 + dx; dx==3 is a zero pad column
#define EQ_SCALE 0.014731391274719739f   // 1/sqrt(512*9) = sqrt(2)/96

// ---------------------------------------------------------------------------
// Pack weights into WMMA B-fragment order (fp32 16x16x4), tap order t = dy*4+dx.
// Bmat[k'][n], k' = ci*12 + t (dx<3 real, dx==3 zero for ALL n), n = out ch
// (n<3 real else 0), scaled. Pair-interleaved: float2 element (p*16+n) =
// {Bmat[2p][n], Bmat[2p+1][n]} so one b64 fills both B VGPRs of a lane.
// 6144*16 floats = 393216 bytes in d_ws.
// ---------------------------------------------------------------------------
__global__ void prep_weight_kernel(const float* __restrict__ w,
                                   float* __restrict__ bp) {
  int tid = blockIdx.x * blockDim.x + threadIdx.x;     // 6144*16 threads
  if (tid >= 6144 * 16) return;
  int n  = tid & 15;
  int kp = tid >> 4;
  int ci = kp / K_PAD;
  int t  = kp - ci * K_PAD;
  int dy = t >> 2, dx = t & 3;
  float val = 0.0f;
  if (dx < 3 && n < 3)
    val = w[n * (C_IN * K_RAW) + ci * K_RAW + dy * 3 + dx] * EQ_SCALE;
  int p = kp >> 1, par = kp & 1;
  bp[(p * 16 + n) * 2 + par] = val;
}

// ---------------------------------------------------------------------------
// FAST kernel: interior tiles (y in [1,126], x0 in {16..96}); wave-uniform tile
// coords forced scalar via readfirstlane so all loads take the SADDR + constant
// per-lane voffset + immediate form, with one scalar base bump per ci.
// One wave = TWO 16-pixel tiles sharing B fragments; one unaligned b64 per
// A-pair (x-contiguous taps); dx==3 pad column of B is zero so the hi-half's
// .y element needs no masking. No bounds checks anywhere.
// ---------------------------------------------------------------------------
__global__ void conv_wmma_fast_kernel(const float* __restrict__ in,
                                      const float* __restrict__ bpack,
                                      float* __restrict__ out) {
  const int lane = threadIdx.x & 31;
  const int wid  = __builtin_amdgcn_readfirstlane(blockIdx.x * 4 + (threadIdx.x >> 5));
  const int b    = wid / 378;                    // 126 rows * 3 pairs per batch
  const int r    = wid - b * 378;
  const int y    = 1 + r / 3;
  const int pi   = r - (r / 3) * 3;
  const int x0A  = 16 + pi * 32;

  const int M    = lane & 15;
  const int hi   = lane >> 4;
  const int xoff = M + hi * 2 - 1;               // [-1, 17]

  // per-lane 32-bit element offsets; `sub` handled by immediate (+128, +256)
  const int offA = (y - 1) * HW_ + x0A + xoff;   // min 15 (y=1,x0=16,xoff=-1)
  const int offB = offA + 16;
  const int bidx = hi * 16 + M;                  // B fragment lane offset

  const float* pin = in + (size_t)b * C_IN * (HW_ * HW_);   // scalar base
  const v2f*   bp  = (const v2f*)bpack;

  v8f cA = {}, cB = {};
  for (int ci = 0; ci < C_IN; ++ci) {
    const float* p  = pin + (size_t)ci * (HW_ * HW_);       // scalar bump
    const v2f*   pb = bp + ci * 96;                         // scalar bump
#pragma unroll
    for (int sub = 0; sub < 3; ++sub) {
      v2f aA, aB;
      __builtin_memcpy(&aA, p + offA + sub * HW_, 8);       // b64, imm = sub*512B
      __builtin_memcpy(&aB, p + offB + sub * HW_, 8);
      v2f bb = pb[sub * 32 + bidx];                         // b64, imm = sub*256B
      cA = __builtin_amdgcn_wmma_f32_16x16x4_f32(
          false, aA, false, bb, (short)0, cA, false, false);
      cB = __builtin_amdgcn_wmma_f32_16x16x4_f32(
          false, aB, false, bb, (short)0, cB, false, false);
    }
  }

  if (M < 3) {
    const int mbase = hi * 8;
    float* poA = out + (((size_t)b * 3 + M) * HW_ + y) * HW_ + x0A + mbase;
    float* poB = poA + 16;
#pragma unroll
    for (int r8 = 0; r8 < 8; ++r8) { poA[r8] = cA[r8]; poB[r8] = cB[r8]; }
  }
}

// ---------------------------------------------------------------------------
// EDGE kernel: tiles touching the border (y==0, y==127, x0==0, x0==112).
// Predicated b32 loads with in-bounds offset-0 fallback. 268 tiles per batch.
// ---------------------------------------------------------------------------
__global__ void conv_wmma_edge_kernel(const float* __restrict__ in,
                                      const float* __restrict__ bpack,
                                      float* __restrict__ out) {
  const int lane = threadIdx.x & 31;
  const int wid  = __builtin_amdgcn_readfirstlane(blockIdx.x * 4 + (threadIdx.x >> 5));
  const int b    = wid / 268;
  const int e    = wid - b * 268;
  int y, xt;
  if (e < 16) { y = (e < 8) ? 0 : 127; xt = e & 7; }
  else        { int e2 = e - 16; y = 1 + (e2 >> 1); xt = (e2 & 1) ? 7 : 0; }
  const int x0 = xt << 4;

  const int M    = lane & 15;
  const int hi   = lane >> 4;
  const int bidx = hi * 16 + M;

  int  off[6];
  bool vld[6];
#pragma unroll
  for (int j = 0; j < 6; ++j) {
    int dy = j >> 1;
    int dx = hi * 2 + (j & 1);
    if (dx < 3) {
      int ym = y + dy - 1;
      int xm = x0 + M + dx - 1;
      vld[j] = (ym >= 0) && (ym < HW_) && (xm >= 0) && (xm < HW_);
      off[j] = vld[j] ? (ym * HW_ + xm) : 0;
    } else { vld[j] = false; off[j] = 0; }        // dx==3 pad
  }

  const float* pin = in + (size_t)b * C_IN * (HW_ * HW_);
  const v2f*   bp  = (const v2f*)bpack;

  v8f c = {};
  for (int ci = 0; ci < C_IN; ++ci) {
    const float* p  = pin + (size_t)ci * (HW_ * HW_);
    const v2f*   pb = bp + ci * 96;
#pragma unroll
    for (int sub = 0; sub < 3; ++sub) {
      float a0 = vld[2 * sub]     ? p[off[2 * sub]]     : 0.0f;
      float a1 = vld[2 * sub + 1] ? p[off[2 * sub + 1]] : 0.0f;
      v2f a; a.x = a0; a.y = a1;
      v2f bb = pb[sub * 32 + bidx];
      c = __builtin_amdgcn_wmma_f32_16x16x4_f32(
          false, a, false, bb, (short)0, c, false, false);
    }
  }

  if (M < 3) {
    const int mbase = hi * 8;
    float* po = out + (((size_t)b * 3 + M) * HW_ + y) * HW_ + x0 + mbase;
#pragma unroll
    for (int r8 = 0; r8 < 8; ++r8) po[r8] = c[r8];
  }
}

// ---------------------------------------------------------------------------
// out += bias + upfirdn2d(skip, up=2, k=[1,3,3,1]^2 /64 * 4).
// Per-axis taps: even coord -> {0.75 @ c/2, 0.25 @ c/2-1},
// odd  coord -> {0.25 @ (c+1)/2, 0.75 @ (c-1)/2}; out-of-range taps dropped.
// ---------------------------------------------------------------------------
__global__ void skip_bias_kernel(const float* __restrict__ skip,
                                 const float* __restrict__ bias,
                                 float* __restrict__ out) {
  int tid = blockIdx.x * blockDim.x + threadIdx.x;
  const int total = NBATCH * 3 * HW_ * HW_;
  if (tid >= total) return;

  int x  = tid & 127;
  int y  = (tid >> 7) & 127;
  int bc = tid >> 14;               // b*3 + co
  int co = bc % 3;

  int y2 = y >> 1; bool ye = !(y & 1);
  int   my0 = ye ? y2     : y2 + 1;  float wy0 = ye ? 0.75f : 0.25f;
  int   my1 = ye ? y2 - 1 : y2;      float wy1 = ye ? 0.25f : 0.75f;
  int x2 = x >> 1; bool xe = !(x & 1);
  int   mx0 = xe ? x2     : x2 + 1;  float wx0 = xe ? 0.75f : 0.25f;
  int   mx1 = xe ? x2 - 1 : x2;      float wx1 = xe ? 0.25f : 0.75f;

  const float* s = skip + (size_t)bc * 64 * 64;
  float acc = bias[co];
  if (my0 >= 0 && my0 < 64) {
    if (mx0 >= 0 && mx0 < 64) acc += wy0 * wx0 * s[my0 * 64 + mx0];
    if (mx1 >= 0 && mx1 < 64) acc += wy0 * wx1 * s[my0 * 64 + mx1];
  }
  if (my1 >= 0 && my1 < 64) {
    if (mx0 >= 0 && mx0 < 64) acc += wy1 * wx0 * s[my1 * 64 + mx0];
    if (mx1 >= 0 && mx1 < 64) acc += wy1 * wx1 * s[my1 * 64 + mx1];
  }
  out[tid] += acc;
}

// ---------------------------------------------------------------------------
extern "C" void kernel_launch(void* const* d_in, const int* in_sizes, int n_in,
                              void* d_out, int out_size, void* d_ws,
                              size_t ws_size, hipStream_t stream) {
  (void)in_sizes; (void)n_in; (void)out_size; (void)ws_size;
  const float* in   = (const float*)d_in[0];   // (8,512,128,128)
  const float* skip = (const float*)d_in[1];   // (8,3,64,64)
  const float* w    = (const float*)d_in[2];   // (3,512,3,3)
  const float* bias = (const float*)d_in[3];   // (3,)
  float* out = (float*)d_out;                  // (8,3,128,128)
  float* bp  = (float*)d_ws;                   // 393216 B weight pack

  prep_weight_kernel<<<dim3((6144 * 16 + 255) / 256), dim3(256), 0, stream>>>(w, bp);
  // interior: 8 * 126 rows * 3 tile-pairs = 3024 waves, 4 waves/block
  conv_wmma_fast_kernel<<<dim3(756), dim3(128), 0, stream>>>(in, bp, out);
  // edges: 8 * 268 tiles = 2144 waves, 4 waves/block
  conv_wmma_edge_kernel<<<dim3(536), dim3(128), 0, stream>>>(in, bp, out);
  skip_bias_kernel<<<dim3((NBATCH * 3 * HW_ * HW_ + 255) / 256), dim3(256), 0, stream>>>(
      skip, bias, out);
}